// GAT_63728724738761
// MI455X (gfx1250) — compile-verified
//
#include <hip/hip_runtime.h>

typedef __attribute__((ext_vector_type(2))) float v2f;
typedef __attribute__((ext_vector_type(8))) float v8f;

// ---------- helpers ----------
__device__ __forceinline__ unsigned fenc(float f) {
    unsigned u = __float_as_uint(f);
    return (u & 0x80000000u) ? ~u : (u | 0x80000000u);   // order-preserving map
}
__device__ __forceinline__ float fdec(unsigned k) {
    unsigned u = (k & 0x80000000u) ? (k ^ 0x80000000u) : ~k;
    return __uint_as_float(u);
}
__device__ __forceinline__ float lrelu(float x) { return x > 0.f ? x : 0.2f * x; }

// ---------- layer 1 GEMM: xh1[N,64] = x[N,128] @ W1[128,64] via WMMA f32 16x16x4 ----------
__global__ __launch_bounds__(128) void k_gemm1(const float* __restrict__ x,
                                               const float* __restrict__ W1,
                                               float* __restrict__ xh1, int n) {
    __shared__ float Wlds[128 * 64];   // 32 KB
    __shared__ float Alds[16 * 128];   // 8 KB
    const int t = threadIdx.x;
    for (int i = t; i < 128 * 64; i += 128) Wlds[i] = W1[i];
    const int rowBase = blockIdx.x * 16;
    for (int i = t; i < 16 * 128; i += 128) {
        int r = i >> 7, c = i & 127;
        int gr = rowBase + r; if (gr >= n) gr = n - 1;
        Alds[i] = x[gr * 128 + c];
    }
    __syncthreads();

    const int wave = t >> 5;
    const int lane = t & 31;
    const int c0   = wave * 16;        // 4 waves cover 64 output cols
    const int hi   = lane >> 4;        // 0: K0/K1 half, 1: K2/K3 half
    const int lr   = lane & 15;

    v8f acc = {};
    for (int k0 = 0; k0 < 128; k0 += 4) {
        const int ka = k0 + hi * 2;
        v2f a, b;
        a.x = Alds[lr * 128 + ka];
        a.y = Alds[lr * 128 + ka + 1];
        b.x = Wlds[ka * 64 + c0 + lr];
        b.y = Wlds[(ka + 1) * 64 + c0 + lr];
        acc = __builtin_amdgcn_wmma_f32_16x16x4_f32(false, a, false, b,
                                                    (short)0, acc, false, false);
    }
    const int col = c0 + lr;
#pragma unroll
    for (int r = 0; r < 8; ++r) {
        const int row = rowBase + hi * 8 + r;
        if (row < n) xh1[row * 64 + col] = acc[r];
    }
}

// ---------- attention logits per node, layer 1 (H=8, C=8) ----------
__global__ void k_al1(const float* __restrict__ xh1,
                      const float* __restrict__ att_src, const float* __restrict__ att_dst,
                      float* __restrict__ alS, float* __restrict__ alD, int n) {
    int i = blockIdx.x * blockDim.x + threadIdx.x;
    if (i >= n) return;
    const float* row = xh1 + (size_t)i * 64;
#pragma unroll
    for (int h = 0; h < 8; ++h) {
        float ss = 0.f, sd = 0.f;
#pragma unroll
        for (int c = 0; c < 8; ++c) {
            float v = row[h * 8 + c];
            ss += v * att_src[h * 8 + c];
            sd += v * att_dst[h * 8 + c];
        }
        alS[i * 8 + h] = ss;
        alD[i * 8 + h] = sd;
    }
}

// ---------- zero fill ----------
__global__ void k_zero(float* __restrict__ p, long long count) {
    long long i = (long long)blockIdx.x * blockDim.x + threadIdx.x;
    if (i < count) p[i] = 0.f;
}

// ---------- layer 1 edge passes: one thread per (edge, head) ----------
__global__ void k_emax1(const long long* __restrict__ ei, int E, int etot,
                        const float* __restrict__ alS, const float* __restrict__ alD,
                        unsigned* __restrict__ mk) {
    int idx = blockIdx.x * blockDim.x + threadIdx.x;
    if (idx >= etot * 8) return;
    int e = idx >> 3, h = idx & 7;
    long long s, d;
    if (e < E) { s = ei[e]; d = ei[E + e]; } else { s = d = e - E; }
    float v = lrelu(alS[s * 8 + h] + alD[d * 8 + h]);
    atomicMax(mk + d * 8 + h, fenc(v));
}

__global__ void k_esum1(const long long* __restrict__ ei, int E, int etot,
                        const float* __restrict__ alS, const float* __restrict__ alD,
                        const unsigned* __restrict__ mk, float* __restrict__ sum) {
    int idx = blockIdx.x * blockDim.x + threadIdx.x;
    if (idx >= etot * 8) return;
    int e = idx >> 3, h = idx & 7;
    long long s, d;
    if (e < E) { s = ei[e]; d = ei[E + e]; } else { s = d = e - E; }
    float v = lrelu(alS[s * 8 + h] + alD[d * 8 + h]);
    float p = __expf(v - fdec(mk[d * 8 + h]));
    atomicAdd(sum + d * 8 + h, p);
}

__global__ void k_eagg1(const long long* __restrict__ ei, int E, int etot,
                        const float* __restrict__ alS, const float* __restrict__ alD,
                        const unsigned* __restrict__ mk, const float* __restrict__ sum,
                        const float* __restrict__ xh1, float* __restrict__ agg) {
    int idx = blockIdx.x * blockDim.x + threadIdx.x;
    if (idx >= etot * 8) return;
    int e = idx >> 3, h = idx & 7;
    long long s, d;
    if (e < E) { s = ei[e]; d = ei[E + e]; } else { s = d = e - E; }
    float v = lrelu(alS[s * 8 + h] + alD[d * 8 + h]);
    float p = __expf(v - fdec(mk[d * 8 + h]));
    float alpha = p / (sum[d * 8 + h] + 1e-16f);
    const float4* xr = (const float4*)(xh1 + (size_t)s * 64 + h * 8);
    float4 x0 = xr[0], x1 = xr[1];
    float* ag = agg + (size_t)d * 64 + h * 8;
    atomicAdd(ag + 0, x0.x * alpha);
    atomicAdd(ag + 1, x0.y * alpha);
    atomicAdd(ag + 2, x0.z * alpha);
    atomicAdd(ag + 3, x0.w * alpha);
    atomicAdd(ag + 4, x1.x * alpha);
    atomicAdd(ag + 5, x1.y * alpha);
    atomicAdd(ag + 6, x1.z * alpha);
    atomicAdd(ag + 7, x1.w * alpha);
}

__global__ void k_fin1(const float* __restrict__ agg, const float* __restrict__ b1,
                       float* __restrict__ hout, long long cnt) {
    long long i = (long long)blockIdx.x * blockDim.x + threadIdx.x;
    if (i >= cnt) return;
    float v = agg[i] + b1[i & 63];
    hout[i] = v > 0.f ? v : 0.f;
}

// ---------- layer 2 GEMM (64x4, skinny) + attention logits ----------
__global__ __launch_bounds__(256) void k_gemm2(const float* __restrict__ h,
                                               const float* __restrict__ W2,
                                               const float* __restrict__ as2,
                                               const float* __restrict__ ad2,
                                               float* __restrict__ xh2,
                                               float* __restrict__ alS, float* __restrict__ alD,
                                               int n) {
    __shared__ float Wl[64 * 4];
    int t = threadIdx.x;
    if (t < 256) Wl[t] = W2[t];
    __syncthreads();
    int i = blockIdx.x * 256 + t;
    if (i >= n) return;
    const float* row = h + (size_t)i * 64;
    float o0 = 0.f, o1 = 0.f, o2 = 0.f, o3 = 0.f;
#pragma unroll 8
    for (int k = 0; k < 64; ++k) {
        float v = row[k];
        o0 += v * Wl[k * 4 + 0];
        o1 += v * Wl[k * 4 + 1];
        o2 += v * Wl[k * 4 + 2];
        o3 += v * Wl[k * 4 + 3];
    }
    xh2[i * 4 + 0] = o0; xh2[i * 4 + 1] = o1;
    xh2[i * 4 + 2] = o2; xh2[i * 4 + 3] = o3;
    alS[i] = o0 * as2[0] + o1 * as2[1] + o2 * as2[2] + o3 * as2[3];
    alD[i] = o0 * ad2[0] + o1 * ad2[1] + o2 * ad2[2] + o3 * ad2[3];
}

// ---------- layer 2 edge passes (H=1, C=4): one thread per edge ----------
__global__ void k_emax2(const long long* __restrict__ ei, int E, int etot,
                        const float* __restrict__ alS, const float* __restrict__ alD,
                        unsigned* __restrict__ mk) {
    int e = blockIdx.x * blockDim.x + threadIdx.x;
    if (e >= etot) return;
    long long s, d;
    if (e < E) { s = ei[e]; d = ei[E + e]; } else { s = d = e - E; }
    atomicMax(mk + d, fenc(lrelu(alS[s] + alD[d])));
}

__global__ void k_esum2(const long long* __restrict__ ei, int E, int etot,
                        const float* __restrict__ alS, const float* __restrict__ alD,
                        const unsigned* __restrict__ mk, float* __restrict__ sum) {
    int e = blockIdx.x * blockDim.x + threadIdx.x;
    if (e >= etot) return;
    long long s, d;
    if (e < E) { s = ei[e]; d = ei[E + e]; } else { s = d = e - E; }
    float v = lrelu(alS[s] + alD[d]);
    atomicAdd(sum + d, __expf(v - fdec(mk[d])));
}

__global__ void k_eagg2(const long long* __restrict__ ei, int E, int etot,
                        const float* __restrict__ alS, const float* __restrict__ alD,
                        const unsigned* __restrict__ mk, const float* __restrict__ sum,
                        const float* __restrict__ xh2, float* __restrict__ agg) {
    int e = blockIdx.x * blockDim.x + threadIdx.x;
    if (e >= etot) return;
    long long s, d;
    if (e < E) { s = ei[e]; d = ei[E + e]; } else { s = d = e - E; }
    float v = lrelu(alS[s] + alD[d]);
    float p = __expf(v - fdec(mk[d]));
    float alpha = p / (sum[d] + 1e-16f);
    float4 xv = *(const float4*)(xh2 + (size_t)s * 4);
    float* ag = agg + (size_t)d * 4;
    atomicAdd(ag + 0, xv.x * alpha);
    atomicAdd(ag + 1, xv.y * alpha);
    atomicAdd(ag + 2, xv.z * alpha);
    atomicAdd(ag + 3, xv.w * alpha);
}

__global__ void k_fin2(const float* __restrict__ agg, const float* __restrict__ b2,
                       float* __restrict__ zout, long long cnt) {
    long long i = (long long)blockIdx.x * blockDim.x + threadIdx.x;
    if (i >= cnt) return;
    zout[i] = agg[i] + b2[i & 3];
}

// ---------- launch ----------
extern "C" void kernel_launch(void* const* d_in, const int* in_sizes, int n_in,
                              void* d_out, int out_size, void* d_ws, size_t ws_size,
                              hipStream_t stream) {
    const float*     x   = (const float*)d_in[0];
    const long long* ei  = (const long long*)d_in[1];   // int64 edge_index [2,E]
    const float*     W1  = (const float*)d_in[2];
    const float*     as1 = (const float*)d_in[3];
    const float*     ad1 = (const float*)d_in[4];
    const float*     b1  = (const float*)d_in[5];
    const float*     W2  = (const float*)d_in[6];
    const float*     as2 = (const float*)d_in[7];
    const float*     ad2 = (const float*)d_in[8];
    const float*     b2  = (const float*)d_in[9];

    const int n    = in_sizes[0] / 128;
    const int E    = in_sizes[1] / 2;
    const int etot = E + n;

    // workspace layout (floats): computed buffers first, then the zeroed region
    float* ws   = (float*)d_ws;
    float* xh1  = ws;                                   // n*64
    float* alS1 = xh1  + (size_t)n * 64;                // n*8
    float* alD1 = alS1 + (size_t)n * 8;                 // n*8
    float* xh2  = alD1 + (size_t)n * 8;                 // n*4
    float* alS2 = xh2  + (size_t)n * 4;                 // n
    float* alD2 = alS2 + (size_t)n;                     // n
    // zeroed region (contiguous): mk1, sum1, agg1, mk2, sum2, agg2
    unsigned* mk1  = (unsigned*)(alD2 + (size_t)n);     // n*8
    float*    sum1 = (float*)mk1 + (size_t)n * 8;       // n*8
    float*    agg1 = sum1 + (size_t)n * 8;              // n*64
    unsigned* mk2  = (unsigned*)(agg1 + (size_t)n * 64);// n
    float*    sum2 = (float*)mk2 + (size_t)n;           // n
    float*    agg2 = sum2 + (size_t)n;                  // n*4
    const long long zcount = (long long)n * (8 + 8 + 64 + 1 + 1 + 4);

    float* hout = (float*)d_out;                        // n*64
    float* zout = hout + (size_t)n * 64;                // n*4

    const int T = 256;
    // layer 1
    k_gemm1<<<(n + 15) / 16, 128, 0, stream>>>(x, W1, xh1, n);
    k_al1<<<(n + T - 1) / T, T, 0, stream>>>(xh1, as1, ad1, alS1, alD1, n);
    k_zero<<<(int)((zcount + T - 1) / T), T, 0, stream>>>((float*)mk1, zcount);
    {
        int work = etot * 8;
        k_emax1<<<(work + T - 1) / T, T, 0, stream>>>(ei, E, etot, alS1, alD1, mk1);
        k_esum1<<<(work + T - 1) / T, T, 0, stream>>>(ei, E, etot, alS1, alD1, mk1, sum1);
        k_eagg1<<<(work + T - 1) / T, T, 0, stream>>>(ei, E, etot, alS1, alD1, mk1, sum1, xh1, agg1);
    }
    k_fin1<<<(int)(((long long)n * 64 + T - 1) / T), T, 0, stream>>>(agg1, b1, hout, (long long)n * 64);
    // layer 2
    k_gemm2<<<(n + T - 1) / T, T, 0, stream>>>(hout, W2, as2, ad2, xh2, alS2, alD2, n);
    k_emax2<<<(etot + T - 1) / T, T, 0, stream>>>(ei, E, etot, alS2, alD2, mk2);
    k_esum2<<<(etot + T - 1) / T, T, 0, stream>>>(ei, E, etot, alS2, alD2, mk2, sum2);
    k_eagg2<<<(etot + T - 1) / T, T, 0, stream>>>(ei, E, etot, alS2, alD2, mk2, sum2, xh2, agg2);
    k_fin2<<<(int)(((long long)n * 4 + T - 1) / T), T, 0, stream>>>(agg2, b2, zout, (long long)n * 4);
}